// AQIP_21895743275770
// MI455X (gfx1250) — compile-verified
//
#include <hip/hip_runtime.h>
#include <hip/hip_bf16.h>

// ---------------------------------------------------------------------------
// CDNA5 / gfx1250. All matmuls on the matrix pipe via V_WMMA_F32_16X16X4_F32
// (fp32 in/out): the workload (~60 GFLOP vs ~1.5 GB of HBM traffic at
// 23.3 TB/s) is memory/latency bound, so we keep full fp32 precision.
//
// WMMA fragment layouts (ISA 7.12.2, wave32):
//   A (16x4 f32):  lane l: m = l&15, half = l>>4;  a[j] = A[m][2*half + j]
//   B (4x16 f32):  lane l: n = l&15;               b[j] = B[j + 2*half][n]
//   C/D (16x16):   lane l: n = l&15;               c[v] = C[v + 8*half][n]
//
// B-matrices are kept "pair-interleaved" (Wp[k/2][n][2] = W[k][n]) so each
// B fragment is a single aligned 8-byte load; ka = k0 + 2*half is always even.
// ---------------------------------------------------------------------------

typedef float v2f __attribute__((ext_vector_type(2)));
typedef float v8f __attribute__((ext_vector_type(8)));

#define BDIM 64
#define TDIM 168
#define NNODE 64
#define FIN 17
#define HID 128
#define BT (BDIM * TDIM)            // 10752
#define NROWS (BT * NNODE)          // 688128
#define GATES 512                   // 4*HID

__device__ __forceinline__ v8f wmma4(v2f a, v2f b, v8f c) {
    return __builtin_amdgcn_wmma_f32_16x16x4_f32(
        /*neg_a=*/false, a, /*neg_b=*/false, b,
        /*c_mod=*/(short)0, c, /*reuse_a=*/false, /*reuse_b=*/false);
}

__device__ __forceinline__ float sigmoidf_(float x) {
    return 1.0f / (1.0f + __expf(-x));
}

// ---------------------------------------------------------------------------
// Prep: WihTp[l][k/2][o][2] = Wih[l][o][k] (transposed + pair-interleaved so
// LSTM B-fragments are one coalesced global_load_b64); bsum = bih + bhh.
// ---------------------------------------------------------------------------
__global__ void prep_kernel(const float* __restrict__ Wih,
                            const float* __restrict__ Whh,
                            const float* __restrict__ bih,
                            const float* __restrict__ bhh,
                            float* __restrict__ WihTp,
                            float* __restrict__ WhhTp,
                            float* __restrict__ bsum) {
    int idx = blockIdx.x * blockDim.x + threadIdx.x;
    const int total = 4 * GATES * HID;   // 262144
    if (idx < total) {
        int l = idx / (GATES * HID);
        int o = (idx / HID) % GATES;
        int k = idx % HID;
        size_t dst = (size_t)l * GATES * HID +
                     (size_t)(k >> 1) * (GATES * 2) + (o << 1) + (k & 1);
        WihTp[dst] = Wih[idx];
        WhhTp[dst] = Whh[idx];
    }
    if (idx < 4 * GATES) bsum[idx] = bih[idx] + bhh[idx];
}

// ---------------------------------------------------------------------------
// GAT linear: H = X @ W (rows x Fin @ Fin x 128), plus per-row
// src = H.a[:128], dst = H.a[128:]. One wave per 16-row tile, 8 waves/block.
// FinT is a compile-time constant: no runtime bounds guards in the K-loop.
// For FinT%4 != 0 the A tile is staged zero-padded in LDS; W is always staged
// zero-padded + pair-interleaved in LDS.
// Safe for X == H in the FinT==128 instantiation (tile reads precede writes,
// single-wave tile ownership).
// ---------------------------------------------------------------------------
template <int FinT>
__global__ __launch_bounds__(256)
void gat_linear_kernel(const float* X,
                       const float* __restrict__ W,     // FinT x 128
                       const float* __restrict__ avec,  // 256
                       float* H,                        // rows x 128
                       float* __restrict__ SRC, float* __restrict__ DST,
                       int nrows) {
    constexpr int FinPad = (FinT + 3) & ~3;
    extern __shared__ float sm[];
    float* sW = sm;                         // FinPad*128, pair-interleaved
    float* sA = sm + FinPad * HID;          // 8 * 16 * FinPad (padded path only)

    int tid = threadIdx.x;
    // stage W: Wp[k/2][n][2], zero-padded rows FinT..FinPad-1
    for (int i = tid; i < FinPad * HID; i += 256) {
        int k = i >> 7;
        int n = i & 127;
        float w = (k < FinT) ? W[k * HID + n] : 0.0f;
        sW[((k >> 1) << 8) + (n << 1) + (k & 1)] = w;
    }

    int wv = tid >> 5, lane = tid & 31;
    int wave = blockIdx.x * 8 + wv;
    int row0 = wave * 16;
    int nl = lane & 15, half = lane >> 4;

    float* sAw = sA + wv * 16 * FinPad;
    if (FinPad != FinT) {
        // stage this wave's 16 x FinT A-tile, zero-padded to FinPad
        for (int i = lane; i < 16 * FinPad; i += 32) {
            int m = i / FinPad, k = i - m * FinPad;
            sAw[i] = (k < FinT) ? X[(size_t)(row0 + m) * FinT + k] : 0.0f;
        }
    }
    __syncthreads();
    if (row0 >= nrows) return;

    v8f acc[8];
#pragma unroll
    for (int n = 0; n < 8; ++n) acc[n] = (v8f){};

    for (int k0 = 0; k0 < FinPad; k0 += 4) {
        int ka = k0 + 2 * half;           // always even
        v2f a;
        if (FinPad != FinT) {
            a = *(const v2f*)&sAw[nl * FinPad + ka];
        } else {
            a = *(const v2f*)(X + (size_t)(row0 + nl) * FinT + ka);
        }
#pragma unroll
        for (int n = 0; n < 8; ++n) {
            v2f b = *(const v2f*)&sW[((ka >> 1) << 8) + ((n * 16 + nl) << 1)];
            acc[n] = wmma4(a, b, acc[n]);
        }
    }

#pragma unroll
    for (int v = 0; v < 8; ++v) {
        int r = row0 + v + 8 * half;
        float ps = 0.0f, pd = 0.0f;
#pragma unroll
        for (int n = 0; n < 8; ++n) {
            float hv = acc[n][v];
            H[(size_t)r * HID + n * 16 + nl] = hv;
            ps += hv * avec[n * 16 + nl];
            pd += hv * avec[HID + n * 16 + nl];
        }
        // reduce across the 16 lanes sharing this row (masks stay in-half)
        for (int m = 8; m >= 1; m >>= 1) {
            ps += __shfl_xor(ps, m, 32);
            pd += __shfl_xor(pd, m, 32);
        }
        if (nl == 0) { SRC[r] = ps; DST[r] = pd; }
    }
}

// ---------------------------------------------------------------------------
// GAT attention (full, in place): per (b,t) block, attn = softmax(mask(
// leaky(src_i + dst_j))), H[bt] = attn @ H[bt]. 4 waves, one 16-row M-tile
// each. h is staged in LDS pair-interleaved (B-frag = one ds_load_b64), so
// the global in-place write is race-free.
// ---------------------------------------------------------------------------
__global__ __launch_bounds__(128)
void gat_attn_kernel(float* __restrict__ H,
                     const float* __restrict__ SRC,
                     const float* __restrict__ DST,
                     const int* __restrict__ adj) {
    __shared__ float hp_s[NNODE * HID];     // 32 KB, pair-interleaved
    __shared__ float attn_s[NNODE * NNODE]; // 16 KB
    __shared__ float src_s[NNODE], dst_s[NNODE];

    int bt = blockIdx.x;
    size_t base = (size_t)bt * NNODE * HID;
    int tid = threadIdx.x;

    for (int i = tid; i < NNODE * HID; i += 128) {
        int j = i >> 7;       // node row
        int c = i & 127;      // hid col
        hp_s[((j >> 1) << 8) + (c << 1) + (j & 1)] = H[base + i];
    }
    if (tid < NNODE) {
        src_s[tid] = SRC[bt * NNODE + tid];
        dst_s[tid] = DST[bt * NNODE + tid];
    }
    __syncthreads();

    if (tid < NNODE) {
        int i = tid;
        float sv = src_s[i];
        float mx = -1e30f;
        for (int j = 0; j < NNODE; ++j) {
            float e = sv + dst_s[j];
            e = (e >= 0.0f) ? e : 0.2f * e;
            bool m = (adj[i * NNODE + j] > 0) || (i == j);
            e = m ? e : -1.0e9f;
            attn_s[i * NNODE + j] = e;
            mx = fmaxf(mx, e);
        }
        float s = 0.0f;
        for (int j = 0; j < NNODE; ++j) {
            float t = __expf(attn_s[i * NNODE + j] - mx);
            attn_s[i * NNODE + j] = t;
            s += t;
        }
        float inv = 1.0f / s;
        for (int j = 0; j < NNODE; ++j) attn_s[i * NNODE + j] *= inv;
    }
    __syncthreads();

    int wv = tid >> 5, lane = tid & 31;
    int nl = lane & 15, half = lane >> 4;
    int mrow0 = wv * 16;

    v8f acc[8];
#pragma unroll
    for (int n = 0; n < 8; ++n) acc[n] = (v8f){};
    for (int k0 = 0; k0 < NNODE; k0 += 4) {
        int ka = k0 + 2 * half;           // even
        v2f a = *(const v2f*)&attn_s[(mrow0 + nl) * NNODE + ka];
#pragma unroll
        for (int n = 0; n < 8; ++n) {
            v2f b = *(const v2f*)&hp_s[((ka >> 1) << 8) + ((n * 16 + nl) << 1)];
            acc[n] = wmma4(a, b, acc[n]);
        }
    }
#pragma unroll
    for (int v = 0; v < 8; ++v)
#pragma unroll
        for (int n = 0; n < 8; ++n)
            H[base + (size_t)(mrow0 + v + 8 * half) * HID + n * 16 + nl] = acc[n][v];
}

// ---------------------------------------------------------------------------
// GAT layer-2 attention, only row site_idx, writing seq time-major [t][b][c].
// ---------------------------------------------------------------------------
__global__ __launch_bounds__(128)
void gat_site_kernel(const float* __restrict__ H,
                     const float* __restrict__ SRC,
                     const float* __restrict__ DST,
                     const int* __restrict__ adj,
                     const int* __restrict__ site_p,
                     float* __restrict__ seq0) {
    __shared__ float e_s[NNODE];
    __shared__ float p_s[NNODE];
    int bt = blockIdx.x;
    int b = bt / TDIM, t = bt - b * TDIM;
    int site = site_p[0];
    int tid = threadIdx.x;
    size_t base = (size_t)bt * NNODE * HID;

    if (tid < NNODE) {
        int j = tid;
        float e = SRC[bt * NNODE + site] + DST[bt * NNODE + j];
        e = (e >= 0.0f) ? e : 0.2f * e;
        bool m = (adj[site * NNODE + j] > 0) || (site == j);
        e_s[j] = m ? e : -1.0e9f;
    }
    __syncthreads();
    if (tid < NNODE) {
        float mx = -1e30f;
        for (int j = 0; j < NNODE; ++j) mx = fmaxf(mx, e_s[j]);
        p_s[tid] = __expf(e_s[tid] - mx);
    }
    __syncthreads();
    float s = 0.0f;
    for (int j = 0; j < NNODE; ++j) s += p_s[j];
    float inv = 1.0f / s;
    float acc = 0.0f;
    for (int j = 0; j < NNODE; ++j) acc += p_s[j] * H[base + j * HID + tid];
    seq0[(size_t)t * BDIM * HID + b * HID + tid] = acc * inv;
}

// ---------------------------------------------------------------------------
// LSTM layer: persistent single workgroup (8 wave32), loops over T=168 steps.
// gates(64x512) = x_t @ WihT + h @ WhhT + bsum via WMMA; wave w owns hid
// columns [w*16, w*16+16) of all 4 gates -> no cross-wave races on c/h-next.
// Weights are pair-interleaved -> each B fragment is one global_load_b64.
// Dynamic LDS: h double-buffer + c + x_t = 4 * 32 KB = 128 KB (< 320 KB/WGP).
// ---------------------------------------------------------------------------
__global__ __launch_bounds__(256)
void lstm_layer_kernel(const float* __restrict__ seq_in,  // [T][64][128]
                       float* __restrict__ seq_out,       // [T][64][128]
                       const float* __restrict__ WihTp,   // [64][512][2]
                       const float* __restrict__ WhhTp,   // [64][512][2]
                       const float* __restrict__ bsum,    // [512]
                       float* __restrict__ hfin) {        // [64][128]
    extern __shared__ float sm[];
    float* h0  = sm;            // 8192
    float* h1  = sm + 8192;     // 8192
    float* c_s = sm + 16384;    // 8192
    float* x_s = sm + 24576;    // 8192

    int tid = threadIdx.x;
    int wv = tid >> 5, lane = tid & 31;
    int nl = lane & 15, half = lane >> 4;
    const int hn = wv;   // hid tile (0..7)

    for (int i = tid; i < 8192; i += 256) { h0[i] = 0.0f; c_s[i] = 0.0f; }
    __syncthreads();

    float* hcur = h0;
    float* hnxt = h1;

    for (int t = 0; t < TDIM; ++t) {
        const float* xt = seq_in + (size_t)t * BDIM * HID;
        for (int i = tid; i < 8192; i += 256) x_s[i] = xt[i];
        __syncthreads();

        for (int mt = 0; mt < 4; ++mt) {
            v8f acc[4];
#pragma unroll
            for (int q = 0; q < 4; ++q) {
                float bq = bsum[q * HID + hn * 16 + nl];
#pragma unroll
                for (int v = 0; v < 8; ++v) acc[q][v] = bq;
            }
            const int arow = (mt * 16 + nl) * HID;

            // x_t @ WihT
#pragma unroll 8
            for (int k0 = 0; k0 < HID; k0 += 4) {
                int ka = k0 + 2 * half;   // even
                v2f a = *(const v2f*)&x_s[arow + ka];
#pragma unroll
                for (int q = 0; q < 4; ++q) {
                    v2f bf = *(const v2f*)&WihTp[(size_t)(ka >> 1) * (GATES * 2) +
                                                 ((q * HID + hn * 16 + nl) << 1)];
                    acc[q] = wmma4(a, bf, acc[q]);
                }
            }
            // h @ WhhT
#pragma unroll 8
            for (int k0 = 0; k0 < HID; k0 += 4) {
                int ka = k0 + 2 * half;   // even
                v2f a = *(const v2f*)&hcur[arow + ka];
#pragma unroll
                for (int q = 0; q < 4; ++q) {
                    v2f bf = *(const v2f*)&WhhTp[(size_t)(ka >> 1) * (GATES * 2) +
                                                 ((q * HID + hn * 16 + nl) << 1)];
                    acc[q] = wmma4(a, bf, acc[q]);
                }
            }
            // elementwise gates + state update (this wave owns these cols)
#pragma unroll
            for (int v = 0; v < 8; ++v) {
                int row = mt * 16 + v + 8 * half;
                int col = hn * 16 + nl;
                int idx = row * HID + col;
                float iv = sigmoidf_(acc[0][v]);
                float fv = sigmoidf_(acc[1][v]);
                float gv = tanhf(acc[2][v]);
                float ov = sigmoidf_(acc[3][v]);
                float cn = fv * c_s[idx] + iv * gv;
                float hv = ov * tanhf(cn);
                c_s[idx] = cn;
                hnxt[idx] = hv;
                seq_out[(size_t)t * BDIM * HID + idx] = hv;
            }
        }
        __syncthreads();
        float* tp = hcur; hcur = hnxt; hnxt = tp;
    }
    for (int i = tid; i < 8192; i += 256) hfin[i] = hcur[i];
}

// ---------------------------------------------------------------------------
// Head: out[b] = hcat[b] . Wlin + blin
// ---------------------------------------------------------------------------
__global__ void head_kernel(const float* __restrict__ hfin,  // [4][64][128]
                            const float* __restrict__ Wlin,  // [512]
                            const float* __restrict__ blin,
                            float* __restrict__ out) {
    int b = threadIdx.x;
    if (b < BDIM) {
        float s = blin[0];
        for (int l = 0; l < 4; ++l)
            for (int k = 0; k < HID; ++k)
                s += hfin[l * BDIM * HID + b * HID + k] * Wlin[l * HID + k];
        out[b] = s;
    }
}

// ---------------------------------------------------------------------------
extern "C" void kernel_launch(void* const* d_in, const int* in_sizes, int n_in,
                              void* d_out, int out_size, void* d_ws, size_t ws_size,
                              hipStream_t stream) {
    const float* x    = (const float*)d_in[0];
    const int*   adj  = (const int*)d_in[1];
    const float* W1   = (const float*)d_in[2];
    const float* a1   = (const float*)d_in[3];
    const float* W2   = (const float*)d_in[4];
    const float* a2   = (const float*)d_in[5];
    const float* Wih  = (const float*)d_in[6];
    const float* Whh  = (const float*)d_in[7];
    const float* bih  = (const float*)d_in[8];
    const float* bhh  = (const float*)d_in[9];
    const float* Wlin = (const float*)d_in[10];
    const float* blin = (const float*)d_in[11];
    const int*   site = (const int*)d_in[12];
    float* out = (float*)d_out;

    char* ws = (char*)d_ws;
    size_t off = 0;
    float* H     = (float*)(ws + off); off += (size_t)NROWS * HID * 4;   // 352 MB
    float* SRC   = (float*)(ws + off); off += (size_t)NROWS * 4;
    float* DST   = (float*)(ws + off); off += (size_t)NROWS * 4;
    float* seqA  = (float*)(ws + off); off += (size_t)TDIM * BDIM * HID * 4;
    float* seqB  = (float*)(ws + off); off += (size_t)TDIM * BDIM * HID * 4;
    float* WihTp = (float*)(ws + off); off += (size_t)4 * HID * GATES * 4;
    float* WhhTp = (float*)(ws + off); off += (size_t)4 * HID * GATES * 4;
    float* bsum  = (float*)(ws + off); off += (size_t)4 * GATES * 4;
    float* hfin  = (float*)(ws + off); off += (size_t)4 * BDIM * HID * 4;

    // weight prep (transpose to k-major pair-interleaved, bias sum)
    prep_kernel<<<1024, 256, 0, stream>>>(Wih, Whh, bih, bhh, WihTp, WhhTp, bsum);

    // GAT layer 1: linear + attention (in place on H)
    gat_linear_kernel<FIN><<<NROWS / 16 / 8, 256,
        (20 * HID + 8 * 16 * 20) * 4, stream>>>(x, W1, a1, H, SRC, DST, NROWS);
    gat_attn_kernel<<<BT, 128, 0, stream>>>(H, SRC, DST, adj);

    // GAT layer 2: linear (in place), then attention only at site_idx
    gat_linear_kernel<HID><<<NROWS / 16 / 8, 256,
        HID * HID * 4, stream>>>(H, W2, a2, H, SRC, DST, NROWS);
    gat_site_kernel<<<BT, 128, 0, stream>>>(H, SRC, DST, adj, site, seqA);

    // 4 LSTM layers (ping-pong seq buffers), 128 KB dynamic LDS each
    lstm_layer_kernel<<<1, 256, 131072, stream>>>(seqA, seqB, WihTp,                    WhhTp,                    bsum,        hfin);
    lstm_layer_kernel<<<1, 256, 131072, stream>>>(seqB, seqA, WihTp + 1 * HID * GATES,  WhhTp + 1 * HID * GATES,  bsum + 512,  hfin + 8192);
    lstm_layer_kernel<<<1, 256, 131072, stream>>>(seqA, seqB, WihTp + 2 * HID * GATES,  WhhTp + 2 * HID * GATES,  bsum + 1024, hfin + 16384);
    lstm_layer_kernel<<<1, 256, 131072, stream>>>(seqB, seqA, WihTp + 3 * HID * GATES,  WhhTp + 3 * HID * GATES,  bsum + 1536, hfin + 24576);

    // linear head
    head_kernel<<<1, 64, 0, stream>>>(hfin, Wlin, blin, out);

    (void)in_sizes; (void)n_in; (void)out_size; (void)ws_size;
}